// ShortConvolution_7181185319141
// MI455X (gfx1250) — compile-verified
//
#include <hip/hip_runtime.h>
#include <stdint.h>

typedef float f4 __attribute__((ext_vector_type(4)));
typedef int   i4 __attribute__((vector_size(16)));   // matches builtin param type

// Problem constants (fixed by the reference).
static constexpr int Bn    = 4;
static constexpr int Tlen  = 4096;
static constexpr int Dch   = 2048;
static constexpr int Dvec  = Dch / 4;      // 512 float4 columns
static constexpr int TT    = 64;           // t-rows per thread
static constexpr int NCHNK = Tlen / TT;    // 64 chunks
static constexpr int STAGE = 8;            // rows per async stage
static constexpr int NSTG  = TT / STAGE;   // 8 stages per chunk

#define AS1 __attribute__((address_space(1)))
#define AS3 __attribute__((address_space(3)))

#if defined(__has_builtin)
#  if __has_builtin(__builtin_amdgcn_global_load_async_to_lds_b128)
#    define HAVE_ASYNC_BUILTIN 1
#  endif
#  if __has_builtin(__builtin_amdgcn_s_wait_asynccnt)
#    define HAVE_WAIT_BUILTIN 1
#  endif
#endif

// Async copy of 16 bytes/lane: global -> LDS, tracked by ASYNCcnt.
__device__ __forceinline__ void async_ld16(const float* g, float* l) {
#ifdef HAVE_ASYNC_BUILTIN
    __builtin_amdgcn_global_load_async_to_lds_b128(
        (AS1 i4*)(uintptr_t)g, (AS3 i4*)l, 0, 0);
#else
    const uint32_t loff = (uint32_t)(uintptr_t)(AS3 void*)l;
    asm volatile("global_load_async_to_lds_b128 %0, %1, off"
                 :: "v"(loff), "v"(g) : "memory");
#endif
}

__device__ __forceinline__ void wait_async_le8() {
#ifdef HAVE_WAIT_BUILTIN
    __builtin_amdgcn_s_wait_asynccnt(8);
#else
    asm volatile("s_wait_asynccnt 0x8" ::: "memory");
#endif
}

__device__ __forceinline__ void wait_async_0() {
#ifdef HAVE_WAIT_BUILTIN
    __builtin_amdgcn_s_wait_asynccnt(0);
#else
    asm volatile("s_wait_asynccnt 0x0" ::: "memory");
#endif
}

__global__ __launch_bounds__(256)
void shortconv_silu_async(const float* __restrict__ x,
                          const float* __restrict__ w,
                          const float* __restrict__ bias,
                          float* __restrict__ out)
{
    // Double-buffered per-lane staging: 2 x 8 rows x 256 lanes x 16B = 64 KB.
    __shared__ float stg[2][STAGE][256 * 4];

    const int tid   = threadIdx.x;
    const int idx   = blockIdx.x * 256 + tid;
    const int dvec  = idx & (Dvec - 1);       // fastest -> coalesced in D
    const int rest  = idx >> 9;               // / Dvec
    const int b     = rest & (Bn - 1);
    const int chunk = rest >> 2;              // / Bn
    const int t0    = chunk * TT;

    const size_t colBase = (size_t)b * Tlen * Dch + (size_t)dvec * 4;
    const float* __restrict__ xcol = x   + colBase;
    float*       __restrict__ ocol = out + colBase;

    // Weights for channels 4*dvec..4*dvec+3 are 16 contiguous floats.
    const f4* __restrict__ wv = (const f4*)(w + (size_t)dvec * 16);
    const f4 w0 = wv[0], w1 = wv[1], w2 = wv[2], w3 = wv[3];
    const f4 wt0 = { w0[0], w1[0], w2[0], w3[0] };  // tap k=0 -> x[t-3]
    const f4 wt1 = { w0[1], w1[1], w2[1], w3[1] };  // tap k=1 -> x[t-2]
    const f4 wt2 = { w0[2], w1[2], w2[2], w3[2] };  // tap k=2 -> x[t-1]
    const f4 wt3 = { w0[3], w1[3], w2[3], w3[3] };  // tap k=3 -> x[t]
    const f4 bv  = *(const f4*)(bias + (size_t)dvec * 4);

    // gfx1250 prefetch of the head of this thread's stream.
    __builtin_prefetch(xcol + (size_t)t0 * Dch, 0, 0);

    // Causal window (zeros left of t=0); t0==0 is the only sub-3 case.
    f4 xm3 = (f4)0.0f, xm2 = (f4)0.0f, xm1 = (f4)0.0f;
    if (t0 != 0) {
        xm3 = *(const f4*)(xcol + (size_t)(t0 - 3) * Dch);
        xm2 = *(const f4*)(xcol + (size_t)(t0 - 2) * Dch);
        xm1 = *(const f4*)(xcol + (size_t)(t0 - 1) * Dch);
    }

    // Prologue: asynchronously stage rows [t0, t0+STAGE) into buffer 0.
#pragma unroll
    for (int r = 0; r < STAGE; ++r)
        async_ld16(xcol + (size_t)(t0 + r) * Dch, &stg[0][r][tid * 4]);

    for (int jj = 0; jj < NSTG; ++jj) {
        const int cur = jj & 1;
        if (jj + 1 < NSTG) {
            const int nxt = cur ^ 1;
            // Issue next stage before waiting on the current one.
#pragma unroll
            for (int r = 0; r < STAGE; ++r)
                async_ld16(xcol + (size_t)(t0 + (jj + 1) * STAGE + r) * Dch,
                           &stg[nxt][r][tid * 4]);
            wait_async_le8();   // in-order completion: current stage resident
        } else {
            wait_async_0();
        }

        size_t off = (size_t)(t0 + jj * STAGE) * Dch;
#pragma unroll
        for (int r = 0; r < STAGE; ++r) {
            const f4 xc = *(const f4*)&stg[cur][r][tid * 4];   // ds_load_b128
            f4 y = bv + wt0 * xm3 + wt1 * xm2 + wt2 * xm1 + wt3 * xc;
            f4 s;
            s[0] = y[0] / (1.0f + __expf(-y[0]));
            s[1] = y[1] / (1.0f + __expf(-y[1]));
            s[2] = y[2] / (1.0f + __expf(-y[2]));
            s[3] = y[3] / (1.0f + __expf(-y[3]));
            __builtin_nontemporal_store(s, (f4*)(ocol + off));
            xm3 = xm2; xm2 = xm1; xm1 = xc;
            off += (size_t)Dch;
        }
    }
}

extern "C" void kernel_launch(void* const* d_in, const int* in_sizes, int n_in,
                              void* d_out, int out_size, void* d_ws, size_t ws_size,
                              hipStream_t stream) {
    const float* x = (const float*)d_in[0];   // [B, T, D] fp32
    const float* w = (const float*)d_in[1];   // [D, 1, K] fp32
    const float* b = (const float*)d_in[2];   // [D] fp32
    float* out     = (float*)d_out;           // [B, T, D] fp32

    const int total   = Bn * Dvec * NCHNK;    // 131072 threads (exact)
    const int threads = 256;
    const int blocks  = total / threads;      // 512 blocks

    hipLaunchKernelGGL(shortconv_silu_async, dim3(blocks), dim3(threads), 0, stream,
                       x, w, b, out);
}